// MemristiveFullyConnected_36223754174667
// MI455X (gfx1250) — compile-verified
//
#include <hip/hip_runtime.h>

// y[B, NOUT] = x[B, NIN] @ (w_pos - w_neg)[NIN, NOUT] + (b_pos - b_neg)[NOUT]
// (exact algebraic simplification of the memristive differential pair; the
//  G_OFF and k_cond/max_w factors cancel identically)
//
// f32 WMMA path: V_WMMA_F32_16X16X4_F32, D = A(16x4) * B(4x16) + C(16x16)

typedef __attribute__((ext_vector_type(2))) float v2f;
typedef __attribute__((ext_vector_type(8))) float v8f;

#define BDIM 128
#define NIN  1024
#define NOUT 1024

__global__ __launch_bounds__(32)
void memristive_fc_wmma_f32(const float* __restrict__ x,
                            const float* __restrict__ wp,
                            const float* __restrict__ wn,
                            const float* __restrict__ bp,
                            const float* __restrict__ bn,
                            float* __restrict__ y) {
    const int lane = threadIdx.x & 31;
    const int half = lane >> 4;   // 0: lanes 0-15, 1: lanes 16-31
    const int l16  = lane & 15;

    // one wave computes a 16(M) x 64(N) strip: 4 adjacent 16x16 tiles
    const int m_tile  = (int)blockIdx.x >> 4;   // 0..7
    const int n_strip = (int)blockIdx.x & 15;   // 0..15
    const int m0 = m_tile * 16;
    const int n0 = n_strip * 64;

    v8f acc[4] = {};   // 4 tiles x 8 f32 accumulator VGPRs

    // A-matrix 16x4 f32 layout (ISA 7.12.2): lanes 0-15 -> M=lane, K=0,1 in
    // VGPR0,VGPR1; lanes 16-31 -> M=lane-16, K=2,3.
    const float* xrow = x + (m0 + l16) * NIN;

    #pragma unroll 4
    for (int k0 = 0; k0 < NIN; k0 += 4) {
        const int ka = k0 + half * 2;          // this lane's K pair start
        v2f a = *(const v2f*)(xrow + ka);      // X[row][ka], X[row][ka+1] (8B aligned)

        // prefetch next K-block of both weight matrices (global_prefetch_b8)
        if (k0 + 4 < NIN) {
            __builtin_prefetch(wp + (ka + 4) * NOUT + n0 + l16, 0, 3);
            __builtin_prefetch(wn + (ka + 4) * NOUT + n0 + l16, 0, 3);
        }

        #pragma unroll
        for (int t = 0; t < 4; ++t) {
            const int col = n0 + t * 16 + l16;
            // B-matrix 4x16 f32, mirroring C/D striping: lane&15 = N column,
            // VGPR0 holds K=ka, VGPR1 holds K=ka+1 (low half K=0,1; high K=2,3)
            const float* p = wp + ka * NOUT + col;
            const float* n = wn + ka * NOUT + col;
            v2f b;
            b.x = p[0]    - n[0];        // (ka  , col)
            b.y = p[NOUT] - n[NOUT];     // (ka+1, col)

            acc[t] = __builtin_amdgcn_wmma_f32_16x16x4_f32(
                /*neg_a=*/false, a, /*neg_b=*/false, b,
                /*c_mod=*/(short)0, acc[t],
                /*reuse_a=*/false, /*reuse_b=*/false);
        }
    }

    // epilogue: bias diff + store.
    // C/D 16x16 f32 layout: VGPR r, lanes 0-15 -> M=r; lanes 16-31 -> M=8+r;
    // N = lane&15 in both halves.
    #pragma unroll
    for (int t = 0; t < 4; ++t) {
        const int col = n0 + t * 16 + l16;
        const float bias = bp[col] - bn[col];
        #pragma unroll
        for (int r = 0; r < 8; ++r) {
            const int row = m0 + half * 8 + r;
            y[row * NOUT + col] = acc[t][r] + bias;
        }
    }
}

extern "C" void kernel_launch(void* const* d_in, const int* in_sizes, int n_in,
                              void* d_out, int out_size, void* d_ws, size_t ws_size,
                              hipStream_t stream) {
    const float* x  = (const float*)d_in[0];  // [128, 1024]
    const float* wp = (const float*)d_in[1];  // [1024, 1024]
    const float* wn = (const float*)d_in[2];  // [1024, 1024]
    const float* bp = (const float*)d_in[3];  // [1024]
    const float* bn = (const float*)d_in[4];  // [1024]
    float* y = (float*)d_out;                 // [128, 1024]

    // 8 M-tiles x 16 N-strips = 128 waves, one wave (32 threads) per block
    dim3 grid(128), block(32);
    hipLaunchKernelGGL(memristive_fc_wmma_f32, grid, block, 0, stream,
                       x, wp, wn, bp, bn, y);
}